// IFNeuron_13700945674756
// MI455X (gfx1250) — compile-verified
//
#include <hip/hip_runtime.h>
#include <cstdint>
#include <cstddef>

// IF-neuron scan over time, faithful to the reference:
//   s = state + x[b,t,n]; spike = (s < 1) ? -1 : 0;
//   state = (s > 1 || s < 1) ? 0 : s      (NaN -> state stays NaN, spike 0)
//
// Memory-bound (512 MiB traffic ~= 22 us at 23.3 TB/s). CDNA5 path used:
// global_load_async_to_lds_b32 (ASYNCcnt-tracked) triple-buffered staging,
// s_wait_asynccnt pipelining, ds_load_b32 consumption, nontemporal stores.

namespace {
constexpr int B      = 32;
constexpr int T      = 1024;
constexpr int N      = 2048;
constexpr int BLOCK  = 256;               // threads per workgroup == n-tile width
constexpr int CHUNK  = 16;                // timesteps staged per LDS buffer
constexpr int NBUF   = 3;                 // triple buffering (<= 48 async ops in flight, cap 63)
constexpr int NCHUNK = T / CHUNK;         // 64

constexpr uint32_t LDS_K_STRIDE    = BLOCK * 4;           // 1024 B between timesteps in LDS
constexpr uint32_t GLB_K_STRIDE    = N * 4;               // 8192 B between timesteps in HBM
constexpr uint32_t BUF_BYTES       = CHUNK * LDS_K_STRIDE;   // 16 KB per buffer
constexpr uint32_t CHUNK_GLB_BYTES = CHUNK * GLB_K_STRIDE;   // 128 KB of global per chunk
} // namespace

// Issue CHUNK async global->LDS b32 copies (one column of timesteps per lane).
// The instruction's imm offset is added to BOTH the LDS and the global address
// so we
// put the LDS k-stride in the imm and fold the residual global k-stride
// (8192 - 1024 = 7168 B) into the 32-bit voffset VGPR -> 1 VALU add per load.
template <int K>
struct IssueK {
  static __device__ __forceinline__ void run(uint32_t lds, uint32_t voff, uint64_t sbase) {
    asm volatile("global_load_async_to_lds_b32 %0, %1, %2 offset:%3"
                 :
                 : "v"(lds),
                   "v"(voff + (uint32_t)K * (GLB_K_STRIDE - LDS_K_STRIDE)),
                   "s"(sbase),
                   "n"(K * (int)LDS_K_STRIDE)
                 : "memory");
    IssueK<K + 1>::run(lds, voff, sbase);
  }
};
template <>
struct IssueK<CHUNK> {
  static __device__ __forceinline__ void run(uint32_t, uint32_t, uint64_t) {}
};

__global__ __launch_bounds__(BLOCK) void if_neuron_scan_kernel(
    const float* __restrict__ x, float* __restrict__ out) {
  __shared__ float buf[NBUF][CHUNK][BLOCK];

  const int tid   = (int)threadIdx.x;
  const int tiles = N / BLOCK;                     // 8 n-tiles per batch
  const int b     = (int)blockIdx.x / tiles;
  const int nt    = (int)blockIdx.x % tiles;
  const int n     = nt * BLOCK + tid;

  // Uniform 64-bit base of this batch's (T, N) slab -> SGPR pair (GVS mode).
  const uint64_t sbase = (uint64_t)(uintptr_t)(x + (size_t)b * T * N);
  // Per-lane 32-bit byte offset of (t = chunk start, n) within the slab.
  uint32_t voff = (uint32_t)(n * 4);
  // Wave-relative LDS byte address of this lane's column in buffer 0.
  const uint32_t lds0 = (uint32_t)(uintptr_t)(&buf[0][0][tid]);

  float* __restrict__ gout = out + (size_t)b * T * N + n;

  // Prime the pipeline: chunks 0 and 1 in flight.
  IssueK<0>::run(lds0 + 0u * BUF_BYTES, voff, sbase);
  IssueK<0>::run(lds0 + 1u * BUF_BYTES, voff + CHUNK_GLB_BYTES, sbase);
  voff += 2u * CHUNK_GLB_BYTES;

  float state = 0.0f;
  int ibuf = 2;   // buffer receiving chunk c+2
  int pbuf = 0;   // buffer holding chunk c

  for (int c = 0; c < NCHUNK; ++c) {
    if (c < NCHUNK - 2) {
      // Issue chunk c+2 (its buffer was fully consumed at iteration c-1).
      IssueK<0>::run(lds0 + (uint32_t)ibuf * BUF_BYTES, voff, sbase);
      voff += CHUNK_GLB_BYTES;
      ibuf = (ibuf == NBUF - 1) ? 0 : ibuf + 1;
      // <=48 outstanding; async loads complete in order, so <=32 left means
      // chunk c has fully landed in LDS.
      asm volatile("s_wait_asynccnt 32" ::: "memory");
    } else if (c == NCHUNK - 2) {
      asm volatile("s_wait_asynccnt 16" ::: "memory");
    } else {
      asm volatile("s_wait_asynccnt 0" ::: "memory");
    }

    // Each lane reads back exactly the LDS words its own async ops wrote:
    // no workgroup barrier required.
    const float* lb = &buf[pbuf][0][tid];
    float* po = gout + (size_t)c * CHUNK * N;
#pragma unroll
    for (int k = 0; k < CHUNK; ++k) {
      float s  = state + lb[(size_t)k * BLOCK];          // ds_load_b32
      bool neg = s < 1.0f;
      bool pos = s > 1.0f;
      __builtin_nontemporal_store(neg ? -1.0f : 0.0f, po + (size_t)k * N);
      state = (neg || pos) ? 0.0f : s;                   // keeps NaN like ref
    }
    pbuf = (pbuf == NBUF - 1) ? 0 : pbuf + 1;
  }
}

extern "C" void kernel_launch(void* const* d_in, const int* in_sizes, int n_in,
                              void* d_out, int out_size, void* d_ws, size_t ws_size,
                              hipStream_t stream) {
  (void)in_sizes; (void)n_in; (void)out_size; (void)d_ws; (void)ws_size;
  const float* x  = (const float*)d_in[0];
  float* out      = (float*)d_out;
  const dim3 grid(B * (N / BLOCK));   // 256 workgroups -> 2048 wave32 waves
  const dim3 block(BLOCK);
  hipLaunchKernelGGL(if_neuron_scan_kernel, grid, block, 0, stream, x, out);
}